// GNN_37941741093195
// MI455X (gfx1250) — compile-verified
//
#include <hip/hip_runtime.h>

typedef __attribute__((ext_vector_type(2))) float v2f;
typedef __attribute__((ext_vector_type(4))) float v4f;
typedef __attribute__((ext_vector_type(8))) float v8f;

#define DH 64   // hidden width, fixed by the model

// ---------------------------------------------------------------------------
// utility: zero a float buffer
// ---------------------------------------------------------------------------
__global__ void zero_f32(float* __restrict__ p, long n) {
  long i = (long)blockIdx.x * blockDim.x + threadIdx.x;
  if (i < n) p[i] = 0.0f;
}

// ---------------------------------------------------------------------------
// in-degree count (float, reused by all 3 layers)
// ---------------------------------------------------------------------------
__global__ void count_deg(const long long* __restrict__ dst,
                          float* __restrict__ cnt, int E) {
  int e = blockIdx.x * blockDim.x + threadIdx.x;
  if (e < E) atomicAdd(&cnt[(int)dst[e]], 1.0f);
}

// ---------------------------------------------------------------------------
// Fused dual GEMM with native f32 WMMA:  Yl = X @ Wl,  Yr = X @ Wr
//   X : [N, F] row-major,  Wl/Wr : [F, 64] row-major,  Yl/Yr : [N, 64]
// One wave (32 lanes) per 16-row tile; 4 col-tiles x 2 weights = 8 v8f
// accumulators, A-fragment shared across all 8 WMMAs per K-step.
// grid.x = N/16 (N = 50000 -> exactly 3125 tiles, no tail, EXEC all ones).
// ---------------------------------------------------------------------------
__global__ void gemm_dual_wmma(const float* __restrict__ X, int F,
                               const float* __restrict__ Wl,
                               const float* __restrict__ Wr,
                               float* __restrict__ Yl,
                               float* __restrict__ Yr) {
  const int lane = threadIdx.x;     // 0..31
  const int m    = lane & 15;       // A: row index / B: col index / C: col index
  const int half = lane >> 4;       // A/B: K-half selector / C: row-half selector
  const int m0   = blockIdx.x * 16;

  v8f accl[4] = {};
  v8f accr[4] = {};

  const float* __restrict__ xrow = X + (size_t)(m0 + m) * F;

  for (int k = 0; k < F; k += 4) {
    // A fragment 16x4 f32: a[v] = X[m0+m][k + 2*half + v]  (contiguous -> b64)
    v2f a = *(const v2f*)(xrow + k + 2 * half);

#pragma unroll
    for (int nt = 0; nt < 4; ++nt) {
      const int col = nt * 16 + m;
      const size_t r0 = (size_t)(k + 2 * half) * DH + col;
      // B fragment 4x16 f32: b[v] = W[k + 2*half + v][col]
      v2f bl, br;
      bl.x = Wl[r0];       bl.y = Wl[r0 + DH];
      br.x = Wr[r0];       br.y = Wr[r0 + DH];

      accl[nt] = __builtin_amdgcn_wmma_f32_16x16x4_f32(
          false, a, false, bl, (short)0, accl[nt], false, false);
      accr[nt] = __builtin_amdgcn_wmma_f32_16x16x4_f32(
          false, a, false, br, (short)0, accr[nt], false, false);
    }
  }

  // C/D layout: VGPR v -> row m0 + v + 8*half, lane%16 -> column
#pragma unroll
  for (int nt = 0; nt < 4; ++nt) {
#pragma unroll
    for (int v = 0; v < 8; ++v) {
      const size_t row = (size_t)(m0 + v + 8 * half);
      Yl[row * DH + nt * 16 + m] = accl[nt][v];
      Yr[row * DH + nt * 16 + m] = accr[nt][v];
    }
  }
}

// ---------------------------------------------------------------------------
// Edge scatter: A[dst[e]][j] += Yl[src[e]][j]   (64 f32 atomics / edge)
// One thread per (edge, feature) element; consecutive j -> coalesced loads,
// atomics resolve in L2 (A is 12.8 MB, L2-resident on 192 MB L2).
// ---------------------------------------------------------------------------
__global__ void scatter_add(const long long* __restrict__ src,
                            const long long* __restrict__ dst,
                            const float* __restrict__ Yl,
                            float* __restrict__ A, int E) {
  long t = (long)blockIdx.x * blockDim.x + threadIdx.x;
  if (t >= (long)E * DH) return;
  const int e = (int)(t >> 6);
  const int j = (int)(t & 63);
  const int s = (int)src[e];
  const int d = (int)dst[e];
  atomicAdd(&A[(size_t)d * DH + j], Yl[(size_t)s * DH + j]);
}

// ---------------------------------------------------------------------------
// H = elu(A / max(cnt,1) + bl + Yr)
// ---------------------------------------------------------------------------
__global__ void combine_elu(const float* __restrict__ A,
                            const float* __restrict__ cnt,
                            const float* __restrict__ bl,
                            const float* __restrict__ Yr,
                            float* __restrict__ H, long n) {
  long i = (long)blockIdx.x * blockDim.x + threadIdx.x;
  if (i >= n) return;
  const long node = i >> 6;
  const int  j    = (int)(i & 63);
  float v = A[i] / fmaxf(cnt[node], 1.0f) + bl[j] + Yr[i];
  H[i] = (v > 0.0f) ? v : expm1f(v);
}

// ---------------------------------------------------------------------------
// Per-edge projection: out[e] = [h[src]|h[dst]] @ Wp + bp   (Wp: [128,4])
// Wp rows are exactly float4 -> vectorized; H is L2-resident.
// ---------------------------------------------------------------------------
__global__ void edge_proj(const float* __restrict__ H,
                          const long long* __restrict__ src,
                          const long long* __restrict__ dst,
                          const float* __restrict__ Wp,
                          const float* __restrict__ bp,
                          float* __restrict__ out, int E) {
  int e = blockIdx.x * blockDim.x + threadIdx.x;
  if (e >= E) return;
  const float* hs = H + (size_t)((int)src[e]) * DH;
  const float* hd = H + (size_t)((int)dst[e]) * DH;

  float a0 = bp[0], a1 = bp[1], a2 = bp[2], a3 = bp[3];
#pragma unroll 8
  for (int k = 0; k < DH; ++k) {
    v4f w = *(const v4f*)(Wp + (size_t)k * 4);
    float s = hs[k];
    a0 += s * w.x; a1 += s * w.y; a2 += s * w.z; a3 += s * w.w;
  }
#pragma unroll 8
  for (int k = 0; k < DH; ++k) {
    v4f w = *(const v4f*)(Wp + (size_t)(DH + k) * 4);
    float s = hd[k];
    a0 += s * w.x; a1 += s * w.y; a2 += s * w.z; a3 += s * w.w;
  }
  v4f r; r.x = a0; r.y = a1; r.z = a2; r.w = a3;
  *(v4f*)(out + (size_t)e * 4) = r;
}

// ---------------------------------------------------------------------------
// Host-side launcher
// ---------------------------------------------------------------------------
extern "C" void kernel_launch(void* const* d_in, const int* in_sizes, int n_in,
                              void* d_out, int out_size, void* d_ws, size_t ws_size,
                              hipStream_t stream) {
  const float*     x   = (const float*)d_in[0];        // [N,128]
  const long long* ei  = (const long long*)d_in[1];    // [2,E] int64
  const float* Wl1 = (const float*)d_in[2];
  const float* bl1 = (const float*)d_in[3];
  const float* Wr1 = (const float*)d_in[4];
  const float* Wl2 = (const float*)d_in[5];
  const float* bl2 = (const float*)d_in[6];
  const float* Wr2 = (const float*)d_in[7];
  const float* Wl3 = (const float*)d_in[8];
  const float* bl3 = (const float*)d_in[9];
  const float* Wr3 = (const float*)d_in[10];
  const float* Wp  = (const float*)d_in[11];
  const float* bp  = (const float*)d_in[12];
  float* out = (float*)d_out;

  const int N = in_sizes[0] / 128;       // 50000
  const int E = in_sizes[1] / 2;         // 800000
  const long NF = (long)N * DH;          // 3.2M

  const long long* src = ei;
  const long long* dst = ei + E;

  // workspace layout (floats): cnt | A | Yl | Yr | Ha | Hb
  float* cnt = (float*)d_ws;
  float* A   = cnt + N;
  float* Yl  = A   + NF;
  float* Yr  = Yl  + NF;
  float* Ha  = Yr  + NF;
  float* Hb  = Ha  + NF;

  const int ZB = 256;
  dim3 b256(256);

  // degree counts (once)
  zero_f32<<<(N + ZB - 1) / ZB, b256, 0, stream>>>(cnt, N);
  count_deg<<<(E + ZB - 1) / ZB, b256, 0, stream>>>(dst, cnt, E);

  const int mTiles = N / 16;                         // 3125, exact
  const long scatN = (long)E * DH;                   // 51.2M
  const int  scatB = (int)((scatN + ZB - 1) / ZB);
  const int  nfB   = (int)((NF + ZB - 1) / ZB);

  // ---- layer 1: X[128] -> Ha[64] ----
  zero_f32<<<nfB, b256, 0, stream>>>(A, NF);
  gemm_dual_wmma<<<mTiles, dim3(32), 0, stream>>>(x, 128, Wl1, Wr1, Yl, Yr);
  scatter_add<<<scatB, b256, 0, stream>>>(src, dst, Yl, A, E);
  combine_elu<<<nfB, b256, 0, stream>>>(A, cnt, bl1, Yr, Ha, NF);

  // ---- layer 2: Ha -> Hb ----
  zero_f32<<<nfB, b256, 0, stream>>>(A, NF);
  gemm_dual_wmma<<<mTiles, dim3(32), 0, stream>>>(Ha, DH, Wl2, Wr2, Yl, Yr);
  scatter_add<<<scatB, b256, 0, stream>>>(src, dst, Yl, A, E);
  combine_elu<<<nfB, b256, 0, stream>>>(A, cnt, bl2, Yr, Hb, NF);

  // ---- layer 3: Hb -> Ha ----
  zero_f32<<<nfB, b256, 0, stream>>>(A, NF);
  gemm_dual_wmma<<<mTiles, dim3(32), 0, stream>>>(Hb, DH, Wl3, Wr3, Yl, Yr);
  scatter_add<<<scatB, b256, 0, stream>>>(src, dst, Yl, A, E);
  combine_elu<<<nfB, b256, 0, stream>>>(A, cnt, bl3, Yr, Ha, NF);

  // ---- edge projection ----
  edge_proj<<<(E + ZB - 1) / ZB, b256, 0, stream>>>(Ha, src, dst, Wp, bp, out, E);
}